// KLDivergenceProbLoss_46059229283271
// MI455X (gfx1250) — compile-verified
//
#include <hip/hip_runtime.h>
#include <hip/hip_bf16.h>
#include <stdint.h>

// ---------------------------------------------------------------------------
// KL divergence between Gaussian soft-histograms (64 bins) of pred / target.
// Strategy (MI455X / gfx1250, wave32):
//  * windowed closed-form Gaussian weights: 3 x v_exp_f32 + 13 ds_add_f32
//    per element instead of 64 exps  (~10x less transcendental work)
//  * input staged HBM -> LDS with global_load_async_to_lds_b128 (ASYNCcnt),
//    double buffered, wave-private (no workgroup barriers in steady state)
//  * finalize: one wave; 64-wide sums done with v_wmma_f32_16x16x4_f32
//    (A = ones 16x4, B = 4x16 data) + 4 shfl_xor steps.
// ---------------------------------------------------------------------------

#define N_BINS 64
#define EPSF   1e-8f
#define WEIGHTF 0.1f

typedef __attribute__((ext_vector_type(2))) float v2f;
typedef __attribute__((ext_vector_type(8))) float v8f;

// ---- ordered-uint encoding for float atomic min/max --------------------------
__device__ __forceinline__ unsigned fkey(float f) {
  unsigned u = __float_as_uint(f);
  return (u & 0x80000000u) ? ~u : (u | 0x80000000u);
}
__device__ __forceinline__ float funkey(unsigned k) {
  return (k & 0x80000000u) ? __uint_as_float(k & 0x7fffffffu)
                           : __uint_as_float(~k);
}

// ---- low 32 bits of a flat pointer to __shared__ == LDS byte offset ----------
__device__ __forceinline__ unsigned lds_offset(const void* p) {
  return (unsigned)(size_t)p;
}

// =============================================================================
// Kernel 0: init workspace (min/max keys + 8x64 histogram accumulators)
// =============================================================================
__global__ void kl_init_kernel(unsigned* mink, unsigned* maxk, float* hist) {
  int i = threadIdx.x;            // 512 threads
  if (i < 4) { mink[i] = 0xFFFFFFFFu; maxk[i] = 0u; }
  hist[i] = 0.0f;                 // 8 rows * 64 bins = 512
}

// =============================================================================
// Kernel 1: per-row min/max of target
// =============================================================================
__global__ void kl_minmax_kernel(const float* __restrict__ target,
                                 unsigned* mink, unsigned* maxk, int N) {
  const int b = blockIdx.y;
  const float* row = target + (size_t)b * (size_t)N;
  float lmin =  __int_as_float(0x7f800000);   // +inf
  float lmax = -__int_as_float(0x7f800000);   // -inf

  const int stride = gridDim.x * blockDim.x * 4;
  int i = (blockIdx.x * blockDim.x + threadIdx.x) * 4;
  for (; i + 3 < N; i += stride) {
    float4 v = *(const float4*)(row + i);
    lmin = fminf(lmin, fminf(fminf(v.x, v.y), fminf(v.z, v.w)));
    lmax = fmaxf(lmax, fmaxf(fmaxf(v.x, v.y), fmaxf(v.z, v.w)));
  }
  for (; i < N; ++i) { float v = row[i]; lmin = fminf(lmin, v); lmax = fmaxf(lmax, v); }

  #pragma unroll
  for (int m = 16; m > 0; m >>= 1) {
    lmin = fminf(lmin, __shfl_xor(lmin, m, 32));
    lmax = fmaxf(lmax, __shfl_xor(lmax, m, 32));
  }
  if ((threadIdx.x & 31) == 0) {
    atomicMin(&mink[b], fkey(lmin));
    atomicMax(&maxk[b], fkey(lmax));
  }
}

// =============================================================================
// Kernel 2: windowed Gaussian soft-histogram with async-to-LDS staging
//   grid = (tiles_x, B, 2)   z==0 -> pred, z==1 -> target
// =============================================================================
__global__ __launch_bounds__(256)
void kl_hist_kernel(const float* __restrict__ pred,
                    const float* __restrict__ target,
                    const unsigned* __restrict__ mink,
                    const unsigned* __restrict__ maxk,
                    float* __restrict__ hist, int N) {
  const int b = blockIdx.y;
  const int t = blockIdx.z;
  const float* src = (t ? target : pred) + (size_t)b * (size_t)N;

  const float vmin  = funkey(mink[b]);
  const float vmax  = funkey(maxk[b]);
  const float scale = 64.0f / (vmax - vmin + EPSF);

  __shared__ float sh_hist[N_BINS];
  __shared__ float stage[8][2][128];     // 8 waves x 2 buffers x 32 lanes x 4 f32

  for (int i = threadIdx.x; i < N_BINS; i += blockDim.x) sh_hist[i] = 0.0f;
  __syncthreads();

  // exp(-j*j/2) for j=1..6
  const float G1 = 0.60653066f, G2 = 0.13533528f, G3 = 0.011108997f,
              G4 = 3.3546263e-4f, G5 = 3.7266532e-6f, G6 = 1.5229980e-8f;

  auto addb = [&](int k, float w) {
    if ((unsigned)k < 64u) atomicAdd(&sh_hist[k], w);   // ds_add_f32
  };
  auto process = [&](float x) {
    float u   = fmaf(x - vmin, scale, -0.5f);  // bin-unit coordinate
    float k0f = rintf(u);
    float fr  = u - k0f;                       // [-0.5, 0.5]
    int   k0  = (int)k0f;
    float tp  = __expf(fr);                    // v_exp_f32
    float tn  = __expf(-fr);
    float a   = __expf(-0.5f * fr * fr);
    addb(k0, a);
    float mp = a * tp, mm = a * tn;
    addb(k0 + 1, mp * G1); addb(k0 - 1, mm * G1); mp *= tp; mm *= tn;
    addb(k0 + 2, mp * G2); addb(k0 - 2, mm * G2); mp *= tp; mm *= tn;
    addb(k0 + 3, mp * G3); addb(k0 - 3, mm * G3); mp *= tp; mm *= tn;
    addb(k0 + 4, mp * G4); addb(k0 - 4, mm * G4); mp *= tp; mm *= tn;
    addb(k0 + 5, mp * G5); addb(k0 - 5, mm * G5); mp *= tp; mm *= tn;
    addb(k0 + 6, mp * G6); addb(k0 - 6, mm * G6);
  };

  const int wid  = threadIdx.x >> 5;
  const int lane = threadIdx.x & 31;
  const int tilesTotal = N >> 10;            // 1024 elements per block-tile

  // wave-private async staging: each wave issues & consumes only its own 512B
  auto issue = [&](int tl, int buf) {
    unsigned lofs = lds_offset(&stage[wid][buf][lane * 4]);
    unsigned long long ga =
        (unsigned long long)(const void*)(src + ((size_t)tl << 10) + wid * 128 + lane * 4);
    asm volatile("global_load_async_to_lds_b128 %0, %1, off"
                 :: "v"(lofs), "v"(ga) : "memory");
  };

  int tile = blockIdx.x;
  if (tile < tilesTotal) issue(tile, 0);
  int buf = 0;
  for (; tile < tilesTotal; tile += gridDim.x) {
    const int nt = tile + gridDim.x;
    if (nt < tilesTotal) {
      issue(nt, buf ^ 1);
      asm volatile("s_wait_asynccnt 0x1" ::: "memory");
    } else {
      asm volatile("s_wait_asynccnt 0x0" ::: "memory");
    }
    float4 v = *(const float4*)&stage[wid][buf][lane * 4];   // ds_load_b128
    process(v.x); process(v.y); process(v.z); process(v.w);
    buf ^= 1;
  }

  // tail (N not multiple of 1024) -- plain global loads
  for (int i = (tilesTotal << 10) + blockIdx.x * blockDim.x + threadIdx.x;
       i < N; i += gridDim.x * blockDim.x)
    process(src[i]);

  __syncthreads();
  for (int i = threadIdx.x; i < N_BINS; i += blockDim.x)
    atomicAdd(&hist[(b * 2 + t) * N_BINS + i], sh_hist[i]);  // global_atomic_add_f32
}

// =============================================================================
// Kernel 3: finalize. One wave. 64-wide sums via v_wmma_f32_16x16x4_f32.
//   Lane l (l<16) holds bins {l, l+16}; lane l (>=16) holds {l+16, l+32}.
//   That is exactly the 4x16 f32 B-matrix layout; A = ones(16x4) gives
//   C[m][n] = sum_k B[k][n]; 4 shfl_xor steps finish the 16 column sums.
// =============================================================================
__device__ __forceinline__ float wave_sum64(float v0, float v1) {
#if __has_builtin(__builtin_amdgcn_wmma_f32_16x16x4_f32)
  v2f a; a[0] = 1.0f; a[1] = 1.0f;
  v2f bm; bm[0] = v0; bm[1] = v1;
  v8f c = {};
  c = __builtin_amdgcn_wmma_f32_16x16x4_f32(false, a, false, bm,
                                            (short)0, c, false, false);
  float p = c[0];
  p += __shfl_xor(p, 1, 32);
  p += __shfl_xor(p, 2, 32);
  p += __shfl_xor(p, 4, 32);
  p += __shfl_xor(p, 8, 32);
  return p;
#else
  float p = v0 + v1;                 // lanes cover all 64 bins exactly once
  #pragma unroll
  for (int m = 16; m > 0; m >>= 1) p += __shfl_xor(p, m, 32);
  return p;
#endif
}

__global__ void kl_finalize_kernel(const float* __restrict__ hist,
                                   float* __restrict__ out) {
  const unsigned lane = threadIdx.x;                 // exactly 32 threads
  const unsigned i0 = (lane < 16u) ? lane : lane + 16u;
  const unsigned i1 = i0 + 16u;

  float acc = 0.0f;
  for (int b = 0; b < 4; ++b) {
    const float* ph = hist + (b * 2 + 0) * N_BINS;
    const float* th = hist + (b * 2 + 1) * N_BINS;
    float p0 = ph[i0], p1 = ph[i1];
    float t0 = th[i0], t1 = th[i1];
    float ps = wave_sum64(p0, p1);
    float ts = wave_sum64(t0, t1);
    float ip = 1.0f / (ps + EPSF);
    float it = 1.0f / (ts + EPSF);
    float pp0 = p0 * ip, pp1 = p1 * ip;
    float tp0 = t0 * it, tp1 = t1 * it;
    float kl0 = tp0 * (__logf(tp0 + EPSF) - __logf(pp0 + EPSF));
    float kl1 = tp1 * (__logf(tp1 + EPSF) - __logf(pp1 + EPSF));
    acc += wave_sum64(kl0, kl1);
  }
  if (lane == 0) out[0] = WEIGHTF * acc * 0.25f;     // mean over B=4
}

// =============================================================================
extern "C" void kernel_launch(void* const* d_in, const int* in_sizes, int n_in,
                              void* d_out, int out_size, void* d_ws, size_t ws_size,
                              hipStream_t stream) {
  const float* pred   = (const float*)d_in[0];
  const float* target = (const float*)d_in[1];
  const int B = 4;
  const int N = in_sizes[0] / B;            // 1048576

  unsigned* mink = (unsigned*)d_ws;         // 4
  unsigned* maxk = mink + 4;                // 4
  float*    hist = (float*)(mink + 8);      // 8 rows x 64 bins

  kl_init_kernel<<<1, 512, 0, stream>>>(mink, maxk, hist);
  kl_minmax_kernel<<<dim3(64, B), 256, 0, stream>>>(target, mink, maxk, N);
  kl_hist_kernel<<<dim3(128, B, 2), 256, 0, stream>>>(pred, target, mink, maxk, hist, N);
  kl_finalize_kernel<<<1, 32, 0, stream>>>(hist, (float*)d_out);
}